// MultiHeadAttention_57836029608550
// MI455X (gfx1250) — compile-verified
//
#include <hip/hip_runtime.h>
#include <hip/hip_bf16.h>

typedef __attribute__((ext_vector_type(2))) float v2f;
typedef __attribute__((ext_vector_type(8))) float v8f;

// ---------------------------------------------------------------------------
// Node projection GEMM: C[M x 128] = A[M x 128] @ W[128 x 128]   (W is [in,out])
// grid.x = M/16 (16-row tile), block = 256 (8 waves); wave w owns N-cols [16w,16w+16)
// A-tile staged once to LDS (coalesced b128), WMMA A-frags fed from LDS.
// WMMA f32 16x16x4, K chained 32x.
// A-frag (16x4 f32): lane<16 -> M=lane, {K=kb,kb+1}; lane>=16 -> M=lane-16, {K=kb+2,kb+3}
// B-frag (4x16 f32): lane<16 -> N=lane, {K=kb,kb+1}; lane>=16 -> N=lane-16, {K=kb+2,kb+3}
// C/D (16x16 f32):  VGPR r: lane<16 -> (M=r, N=lane); lane>=16 -> (M=r+8, N=lane-16)
// ---------------------------------------------------------------------------
__global__ __launch_bounds__(256) void gemm_node_kernel(
    const float* __restrict__ A, const float* __restrict__ W,
    float* __restrict__ C, int M)
{
    __shared__ float Als[16 * 128];

    const int m0  = blockIdx.x * 16;
    if (m0 >= M) return;
    const int tid = threadIdx.x;

    // stage 16x128 A-tile to LDS: 512 float4, coalesced
    {
        const float4* Ag  = (const float4*)(A + (long)m0 * 128);
        float4*       As4 = (float4*)Als;
        As4[tid]       = Ag[tid];
        As4[tid + 256] = Ag[tid + 256];
    }
    __syncthreads();

    const int wave = tid >> 5;
    const int lane = tid & 31;
    const int n0   = wave * 16;
    const int half = lane >> 4;   // 0 | 1
    const int l16  = lane & 15;

    v8f acc = {};
    const float* Arow = Als + l16 * 128;
    #pragma unroll 4
    for (int kb = 0; kb < 128; kb += 4) {
        const int kr = kb + half * 2;
        v2f a = *(const v2f*)(Arow + kr);          // A[m, kr], A[m, kr+1]  (LDS)
        v2f b;
        b.x = W[(long)kr * 128 + n0 + l16];        // W[kr,   n]
        b.y = W[(long)(kr + 1) * 128 + n0 + l16];  // W[kr+1, n]
        acc = __builtin_amdgcn_wmma_f32_16x16x4_f32(
                  false, a, false, b, (short)0, acc, false, false);
    }
    float* Cb = C + (long)m0 * 128 + n0 + l16;
    #pragma unroll
    for (int r = 0; r < 8; ++r) {
        const int mrow = r + half * 8;
        Cb[(long)mrow * 128] = acc[r];
    }
}

// ---------------------------------------------------------------------------
// Fused edge kernel: one block = 16 edges.
//  phase 0: stage src/dst indices and the 16x128 e-tile into LDS (single touch
//           of the streamed 410MB e array).
//  phase 1: gather Qn[dst], Vn[src] to LDS; WMMA Ee = e_tile @ Wk (A from LDS,
//           8 waves x 16-col tiles); Ke = Ee + Kn[src] -> LDS.
//  phase 2: wave w handles edges {2w, 2w+1}:
//           s[h,g] = <Q[h,:], Ke[g,:]> / 4, leaky-relu, softmax over g,
//           He[h,:] = sum_g A[h,g] * V[g,:], atomicAdd into out[dst].
// ---------------------------------------------------------------------------
__global__ __launch_bounds__(256) void edge_attn_kernel(
    const float* __restrict__ Earr, const int* __restrict__ src,
    const int* __restrict__ dst,   const float* __restrict__ Wk,
    const float* __restrict__ Qn,  const float* __restrict__ Kn,
    const float* __restrict__ Vn,  float* __restrict__ out, int E)
{
    __shared__ int   sIdx[16];
    __shared__ int   dIdx[16];
    __shared__ float Els[16 * 128];
    __shared__ float Ke [16 * 128];
    __shared__ float Qd [16 * 128];
    __shared__ float Vs [16 * 128];

    const int e0  = blockIdx.x * 16;
    if (e0 >= E) return;
    const int tid = threadIdx.x;

    if (tid < 16) { sIdx[tid] = src[e0 + tid]; dIdx[tid] = dst[e0 + tid]; }
    // stage 16x128 e-tile to LDS: 512 float4, coalesced (streamed, read once)
    {
        const float4* Eg  = (const float4*)(Earr + (long)e0 * 128);
        float4*       Es4 = (float4*)Els;
        Es4[tid]       = Eg[tid];
        Es4[tid + 256] = Eg[tid + 256];
    }
    __syncthreads();

    // gather node-projected Q (by dst) and V (by src) into LDS — coalesced in col
    for (int j = tid; j < 16 * 128; j += 256) {
        const int ei = j >> 7, col = j & 127;
        Qd[j] = Qn[(long)dIdx[ei] * 128 + col];
        Vs[j] = Vn[(long)sIdx[ei] * 128 + col];
    }

    // WMMA: 16x128 tile of e @ Wk, wave w -> cols [16w, 16w+16), A from LDS
    const int wave = tid >> 5;
    const int lane = tid & 31;
    const int n0   = wave * 16;
    const int half = lane >> 4;
    const int l16  = lane & 15;

    v8f acc = {};
    const float* Arow = Els + l16 * 128;
    #pragma unroll 4
    for (int kb = 0; kb < 128; kb += 4) {
        const int kr = kb + half * 2;
        v2f a = *(const v2f*)(Arow + kr);
        v2f b;
        b.x = Wk[(long)kr * 128 + n0 + l16];
        b.y = Wk[(long)(kr + 1) * 128 + n0 + l16];
        acc = __builtin_amdgcn_wmma_f32_16x16x4_f32(
                  false, a, false, b, (short)0, acc, false, false);
    }
    #pragma unroll
    for (int r = 0; r < 8; ++r) {
        const int mrow = r + half * 8;                 // edge-in-tile
        Ke[mrow * 128 + n0 + l16] =
            acc[r] + Kn[(long)sIdx[mrow] * 128 + n0 + l16];
    }
    __syncthreads();

    // ---- per-edge attention: wave handles edges 2w and 2w+1 ----
    const float inv_sqrt_dk = 0.25f;   // 1/sqrt(16)
    const int h   = lane >> 2;         // head 0..7
    const int q4  = lane & 3;          // quarter within head group
    const int g0  = q4 * 2;            // this lane's pair of g-heads

    #pragma unroll
    for (int t = 0; t < 2; ++t) {
        const int ei = wave * 2 + t;
        const float* Qrow = Qd + ei * 128 + h * 16;
        const float* K0   = Ke + ei * 128 + g0 * 16;
        const float* K1   = K0 + 16;
        float s0 = 0.f, s1 = 0.f;
        #pragma unroll
        for (int d = 0; d < 16; ++d) {
            const float qv = Qrow[d];
            s0 = fmaf(qv, K0[d], s0);
            s1 = fmaf(qv, K1[d], s1);
        }
        s0 *= inv_sqrt_dk;  s1 *= inv_sqrt_dk;
        s0 = s0 > 0.f ? s0 : 0.01f * s0;   // leaky_relu (jax default slope)
        s1 = s1 > 0.f ? s1 : 0.01f * s1;

        // softmax over g: 8 values live across the aligned group of 4 lanes
        float m = fmaxf(s0, s1);
        m = fmaxf(m, __shfl_xor(m, 1));
        m = fmaxf(m, __shfl_xor(m, 2));
        float x0 = __expf(s0 - m), x1 = __expf(s1 - m);
        float sum = x0 + x1;
        sum += __shfl_xor(sum, 1);
        sum += __shfl_xor(sum, 2);
        const float rinv = 1.f / sum;
        const float a0 = x0 * rinv, a1 = x1 * rinv;

        // He[h, d] for d in [q4*4, q4*4+4): sum_g A[h,g] * V[g,d]
        float he[4] = {0.f, 0.f, 0.f, 0.f};
        const int base = lane & ~3;
        #pragma unroll
        for (int j = 0; j < 4; ++j) {
            const float aa = __shfl(a0, base + j);   // A[h, 2j]
            const float ab = __shfl(a1, base + j);   // A[h, 2j+1]
            const float* Vg0 = Vs + ei * 128 + (2 * j) * 16 + q4 * 4;
            const float* Vg1 = Vg0 + 16;
            #pragma unroll
            for (int d = 0; d < 4; ++d)
                he[d] = fmaf(aa, Vg0[d], fmaf(ab, Vg1[d], he[d]));
        }
        float* ob = out + (long)dIdx[ei] * 128 + h * 16 + q4 * 4;
        #pragma unroll
        for (int d = 0; d < 4; ++d) atomicAdd(ob + d, he[d]);
    }
}

// ---------------------------------------------------------------------------
extern "C" void kernel_launch(void* const* d_in, const int* in_sizes, int n_in,
                              void* d_out, int out_size, void* d_ws, size_t ws_size,
                              hipStream_t stream)
{
    const float* q   = (const float*)d_in[0];
    const float* k   = (const float*)d_in[1];
    const float* v   = (const float*)d_in[2];
    const float* e   = (const float*)d_in[3];
    const int*   src = (const int*)d_in[4];
    const int*   dst = (const int*)d_in[5];
    const float* Wq  = (const float*)d_in[6];
    const float* Wk  = (const float*)d_in[7];
    const float* Wv  = (const float*)d_in[8];

    const int N = in_sizes[0] / 128;   // 50000
    const int E = in_sizes[4];         // 800000
    float* out = (float*)d_out;

    float* Qn = (float*)d_ws;
    float* Kn = Qn + (size_t)N * 128;
    float* Vn = Kn + (size_t)N * 128;

    // segment_sum accumulator must start at zero (capture-safe memset)
    hipMemsetAsync(d_out, 0, (size_t)N * 128 * sizeof(float), stream);

    const int mblocks = (N + 15) / 16;   // N = 50000 = 3125 * 16
    gemm_node_kernel<<<mblocks, 256, 0, stream>>>(q, Wq, Qn, N);
    gemm_node_kernel<<<mblocks, 256, 0, stream>>>(k, Wk, Kn, N);
    gemm_node_kernel<<<mblocks, 256, 0, stream>>>(v, Wv, Vn, N);

    const int eblocks = (E + 15) / 16;   // E = 800000 = 50000 * 16
    edge_attn_kernel<<<eblocks, 256, 0, stream>>>(e, src, dst, Wk, Qn, Kn, Vn, out, E);
}